// VectorSpinModel_78271484002442
// MI455X (gfx1250) — compile-verified
//
#include <hip/hip_runtime.h>

// Problem constants (match reference)
#define N_SPINS 256
#define F_DIM   64
#define B_BATCH 32
#define MAX_IT  30
#define NS_IT   14      // Newton-Schulz iterations (even, so final inverse lands in X)
#define TOL_C   1e-4f

// LDS tile strides (floats). Padded for bank-conflict-free ds reads while
// keeping every row 16B-aligned for global_load_async_to_lds_b128 writes.
#define AS_STRIDE 36    // 64 rows x 32 cols panel, 36*4=144B row pitch (9x16B)
#define BS_STRIDE 72    // 32 rows x 64 cols panel, 72*4=288B row pitch (18x16B)

typedef __attribute__((ext_vector_type(2))) float v2f;
typedef __attribute__((ext_vector_type(8))) float v8f;

// ---------------------------------------------------------------------------
// J = 0.5*(Jraw + Jraw^T) with zero diagonal.  grid(n) x block(n)
// ---------------------------------------------------------------------------
__global__ void prep_J(const float* __restrict__ Jraw, float* __restrict__ J, int n) {
    int i = blockIdx.x, j = threadIdx.x;
    float v = 0.5f * (Jraw[i * n + j] + Jraw[j * n + i]);
    if (i == j) v = 0.0f;
    J[i * n + j] = v;
}

// t = T_INIT, Us = 0, VTs = 0.  grid(B) x block(n)
__global__ void init_state(float* __restrict__ t, float* __restrict__ Us,
                           float* __restrict__ VTs, int n, int mi) {
    int b = blockIdx.x, tid = threadIdx.x;
    t[(long long)b * n + tid] = 4.0f;
    for (int l = 0; l < mi; ++l) {
        Us[((long long)b * n + tid) * mi + l] = 0.0f;
        VTs[((long long)b * mi + l) * n + tid] = 0.0f;
    }
}

// V[b] = diag(t[b]) - J ; rowsum[b][i] = sum_j |V[b][i][j]|.  grid(n,B) x block(n)
__global__ void build_V(const float* __restrict__ J, const float* __restrict__ t,
                        float* __restrict__ V, float* __restrict__ rowsum, int n) {
    int b = blockIdx.y, i = blockIdx.x, j = threadIdx.x;
    float v = ((i == j) ? t[(long long)b * n + i] : 0.0f) - J[i * n + j];
    V[((long long)b * n + i) * n + j] = v;
    __shared__ float s[N_SPINS];
    s[j] = fabsf(v);
    __syncthreads();
    for (int o = N_SPINS / 2; o > 0; o >>= 1) {
        if (j < o) s[j] += s[j + o];
        __syncthreads();
    }
    if (j == 0) rowsum[(long long)b * n + i] = s[0];
}

// alpha[b] = 1 / max_i rowsum[b][i]  (safe Newton-Schulz seed).  grid(B) x block(n)
__global__ void alpha_kernel(const float* __restrict__ rowsum, float* __restrict__ alpha, int n) {
    int b = blockIdx.x, tid = threadIdx.x;
    __shared__ float s[N_SPINS];
    s[tid] = rowsum[(long long)b * n + tid];
    __syncthreads();
    for (int o = N_SPINS / 2; o > 0; o >>= 1) {
        if (tid < o) s[tid] = fmaxf(s[tid], s[tid + o]);
        __syncthreads();
    }
    if (tid == 0) alpha[b] = 1.0f / s[0];
}

// X[b] = alpha[b] * I.  grid(n,B) x block(n)
__global__ void init_X(const float* __restrict__ alpha, float* __restrict__ X, int n) {
    int b = blockIdx.y, i = blockIdx.x, j = threadIdx.x;
    X[((long long)b * n + i) * n + j] = (i == j) ? alpha[b] : 0.0f;
}

// ---------------------------------------------------------------------------
// Newton-Schulz GEMM, specialized 256x256x256, batched over blockIdx.y:
//   mode 0:  C[b] = A[b] @ B[b]
//   mode 1:  C[b] = 2*E[b] - A[b] @ B[b]
// Block = 128 threads (4 waves) -> 64x64 C tile; each wave a 32x32 register
// block (4 accumulators, 2x2 of 16x16 WMMA tiles).
// K consumed in 32-wide panels staged global->LDS with CDNA5 async-tensor
// path (GLOBAL_LOAD_ASYNC_TO_LDS_B128 + s_wait_asynccnt), fragments read
// from LDS (ds_load), MACs via V_WMMA_F32_16X16X4_F32.
// grid((256/64)*(256/64)=16, B) x block(128)
// ---------------------------------------------------------------------------
__global__ void __launch_bounds__(128) gemm_ns(const float* __restrict__ Abase,
                                               const float* __restrict__ Bbase,
                                               float* __restrict__ Cbase,
                                               const float* __restrict__ Ebase,
                                               int mode) {
    __shared__ float As[64 * AS_STRIDE];
    __shared__ float Bs[32 * BS_STRIDE];

    const int n = N_SPINS;
    int tid = threadIdx.x;
    int lane = tid & 31;
    int wave = tid >> 5;
    int half = lane >> 4;
    int r = lane & 15;
    long long bb = blockIdx.y;
    const long long snn = (long long)n * n;
    const float* A = Abase + bb * snn;
    const float* B = Bbase + bb * snn;
    float* C = Cbase + bb * snn;

    int tmB = (blockIdx.x >> 2) << 6;   // block row origin
    int tnB = (blockIdx.x & 3) << 6;    // block col origin
    int mW = (wave >> 1) << 5;          // wave sub-block within 64x64
    int nW = (wave & 1) << 5;

    // Async staging lane maps: each b128 moves 512B per wave.
    // A panel: 64 rows x 128B -> 16 instrs/block (4 rows each), 4 per wave.
    int aRowL = (wave << 4) + (lane >> 3);   // + q*4
    int aChunk = lane & 7;                   // 16B chunk in row
    // B panel: 32 rows x 256B -> 16 instrs/block (2 rows each), 4 per wave.
    int bRowL = (wave << 3) + (lane >> 4);   // + q*2
    int bChunk = lane & 15;

    unsigned asBase = (unsigned)(size_t)(&As[0]);
    unsigned bsBase = (unsigned)(size_t)(&Bs[0]);

    v8f acc00 = {0.f, 0.f, 0.f, 0.f, 0.f, 0.f, 0.f, 0.f};
    v8f acc01 = acc00, acc10 = acc00, acc11 = acc00;

    for (int kb = 0; kb < n; kb += 32) {
        __syncthreads();  // previous compute done before overwriting panels
#pragma unroll
        for (int q = 0; q < 4; ++q) {
            int row = aRowL + (q << 2);
            const float* gp = A + (long long)(tmB + row) * n + kb + (aChunk << 2);
            unsigned lp = asBase + (unsigned)(row * (AS_STRIDE * 4) + (aChunk << 4));
            asm volatile("global_load_async_to_lds_b128 %0, %1, off"
                         :: "v"(lp), "v"(gp) : "memory");
        }
#pragma unroll
        for (int q = 0; q < 4; ++q) {
            int row = bRowL + (q << 1);
            const float* gp = B + (long long)(kb + row) * n + tnB + (bChunk << 2);
            unsigned lp = bsBase + (unsigned)(row * (BS_STRIDE * 4) + (bChunk << 4));
            asm volatile("global_load_async_to_lds_b128 %0, %1, off"
                         :: "v"(lp), "v"(gp) : "memory");
        }
        asm volatile("s_wait_asynccnt 0x0" ::: "memory");
        __syncthreads();

#pragma unroll
        for (int kk = 0; kk < 32; kk += 4) {
            int c0 = kk + (half << 1);
            v2f a0, a1, b0, b1;
            a0.x = As[(mW + r) * AS_STRIDE + c0];
            a0.y = As[(mW + r) * AS_STRIDE + c0 + 1];
            a1.x = As[(mW + 16 + r) * AS_STRIDE + c0];
            a1.y = As[(mW + 16 + r) * AS_STRIDE + c0 + 1];
            b0.x = Bs[c0 * BS_STRIDE + nW + r];
            b0.y = Bs[(c0 + 1) * BS_STRIDE + nW + r];
            b1.x = Bs[c0 * BS_STRIDE + nW + 16 + r];
            b1.y = Bs[(c0 + 1) * BS_STRIDE + nW + 16 + r];
            acc00 = __builtin_amdgcn_wmma_f32_16x16x4_f32(false, a0, false, b0,
                                                          (short)0, acc00, false, false);
            acc01 = __builtin_amdgcn_wmma_f32_16x16x4_f32(false, a0, false, b1,
                                                          (short)0, acc01, false, false);
            acc10 = __builtin_amdgcn_wmma_f32_16x16x4_f32(false, a1, false, b0,
                                                          (short)0, acc10, false, false);
            acc11 = __builtin_amdgcn_wmma_f32_16x16x4_f32(false, a1, false, b1,
                                                          (short)0, acc11, false, false);
        }
    }

    const float* E = Ebase ? (Ebase + bb * snn) : nullptr;
    auto store_tile = [&](v8f acc, int mo, int no) {
#pragma unroll
        for (int j = 0; j < 8; ++j) {
            long long m = tmB + mW + mo + (half << 3) + j;
            long long nn = tnB + nW + no + r;
            float val = acc[j];
            if (mode) val = 2.0f * E[m * n + nn] - val;
            C[m * n + nn] = val;
        }
    };
    store_tile(acc00, 0, 0);
    store_tile(acc01, 0, 16);
    store_tile(acc10, 16, 0);
    store_tile(acc11, 16, 16);
}

// ---------------------------------------------------------------------------
// Generic batched WMMA GEMM (used for A = Vinv @ h, M=256 N=64 K=256):
// one 16x16 tile per wave, direct global fragment loads.
// grid(tiles/4, B) x block(128)
// ---------------------------------------------------------------------------
__global__ void gemm_wmma(const float* __restrict__ Abase, const float* __restrict__ Bbase,
                          float* __restrict__ Cbase,
                          int M, int Nn, int K, int lda, int ldb, int ldc,
                          long long sA, long long sB, long long sC) {
    int lane = threadIdx.x & 31;
    int wave = threadIdx.x >> 5;
    int tile = blockIdx.x * (blockDim.x >> 5) + wave;
    int tilesN = Nn >> 4;
    int tm = (tile / tilesN) << 4;
    int tn = (tile % tilesN) << 4;
    int half = lane >> 4;
    int r = lane & 15;
    long long bb = blockIdx.y;
    const float* A = Abase + bb * sA;
    const float* B = Bbase + bb * sB;
    float* C = Cbase + bb * sC;

    v8f acc = {0.f, 0.f, 0.f, 0.f, 0.f, 0.f, 0.f, 0.f};
    const float* ap = A + (long long)(tm + r) * lda + (half << 1);
    const float* bp = B + (long long)(half << 1) * ldb + tn + r;
    for (int k = 0; k < K; k += 4) {
        v2f a, bf;
        a.x = ap[0];
        a.y = ap[1];
        bf.x = bp[0];
        bf.y = bp[ldb];
        acc = __builtin_amdgcn_wmma_f32_16x16x4_f32(false, a, false, bf,
                                                    (short)0, acc, false, false);
        ap += 4;
        bp += (long long)4 * ldb;
    }
#pragma unroll
    for (int j = 0; j < 8; ++j) {
        long long m = tm + (half << 3) + j;
        C[m * ldc + tn + r] = acc[j];
    }
}

// g[b][i] = 1 - 0.5*Vinv[i][i] - (1/256) * sum_f A[i][f]^2   (Vinv symmetric)
// grid(B) x block(n)
__global__ void grad_kernel(const float* __restrict__ X, const float* __restrict__ Aw,
                            float* __restrict__ g, int n, int f) {
    int b = blockIdx.x, i = threadIdx.x;
    const float* Ar = Aw + ((long long)b * n + i) * f;
    float e = 0.0f;
    for (int q = 0; q < f; ++q) {
        float a = Ar[q];
        e += a * a;
    }
    float d = X[((long long)b * n + i) * n + i];
    g[(long long)b * n + i] = 1.0f - 0.5f * d - (1.0f / 256.0f) * e;
}

// Broyden step, first half: conv test, dx = g - Us@(VTs@g) (masked), t_new = t + dx
// grid(B) x block(n)
__global__ void broyden_pre(const float* __restrict__ g, const float* __restrict__ t,
                            const float* __restrict__ Us, const float* __restrict__ VTs,
                            float* __restrict__ dx, float* __restrict__ tnew,
                            float* __restrict__ convb, int n, int mi) {
    int b = blockIdx.x, tid = threadIdx.x;
    __shared__ float gS[N_SPINS];
    __shared__ float s[N_SPINS];
    __shared__ float w[MAX_IT];
    __shared__ float convS;
    long long base = (long long)b * n + tid;
    float gi = g[base];
    gS[tid] = gi;
    s[tid] = gi * gi;
    __syncthreads();
    for (int o = N_SPINS / 2; o > 0; o >>= 1) {
        if (tid < o) s[tid] += s[tid + o];
        __syncthreads();
    }
    if (tid == 0) {
        float c = (sqrtf(s[0]) < TOL_C) ? 1.0f : 0.0f;
        convS = c;
        convb[b] = c;
    }
    __syncthreads();
    if (tid < mi) {
        const float* vr = VTs + ((long long)b * mi + tid) * n;
        float a = 0.0f;
        for (int q = 0; q < n; ++q) a += vr[q] * gS[q];
        w[tid] = a;
    }
    __syncthreads();
    const float* ur = Us + base * mi;
    float a = 0.0f;
    for (int l = 0; l < mi; ++l) a += ur[l] * w[l];
    float d = gi - a;                 // dx = -apply_Binv(g) = g - Us@(VTs@g)
    if (convS > 0.5f) d = 0.0f;
    dx[base] = d;
    tnew[base] = t[base] + d;
}

// Broyden step, second half: rank-1 update of inverse-Jacobian factors.
// grid(B) x block(n)
__global__ void broyden_post(const float* __restrict__ gnew, float* __restrict__ g,
                             float* __restrict__ t, const float* __restrict__ tnew,
                             const float* __restrict__ dx,
                             float* __restrict__ Us, float* __restrict__ VTs,
                             const float* __restrict__ convb, int k, int n, int mi) {
    int b = blockIdx.x, tid = threadIdx.x;
    __shared__ float dgS[N_SPINS], dxS[N_SPINS], s[N_SPINS];
    __shared__ float w[MAX_IT], w2[MAX_IT];
    bool conv = convb[b] > 0.5f;
    long long base = (long long)b * n + tid;
    float gi = g[base], gni = gnew[base];
    float dgi = gni - gi;
    float dxi = dx[base];
    dgS[tid] = dgi;
    dxS[tid] = dxi;
    __syncthreads();
    if (tid < mi) {
        const float* vr = VTs + ((long long)b * mi + tid) * n;
        float a = 0.0f, a2 = 0.0f;
        for (int q = 0; q < n; ++q) {
            a += vr[q] * dgS[q];                                   // (VTs@dg)_l
            a2 += Us[((long long)b * n + q) * mi + tid] * dxS[q];  // (Us^T dx)_l
        }
        w[tid] = a;
        w2[tid] = a2;
    }
    __syncthreads();
    const float* ur = Us + base * mi;
    float bd = 0.0f, vv = 0.0f;
    for (int l = 0; l < mi; ++l) {
        bd += ur[l] * w[l];
        vv += VTs[((long long)b * mi + l) * n + tid] * w2[l];
    }
    float Binv_dg = -dgi + bd;    // apply_Binv(dg)
    float vT = -dxi + vv;         // B_inv^T dx
    s[tid] = vT * dgi;
    __syncthreads();
    for (int o = N_SPINS / 2; o > 0; o >>= 1) {
        if (tid < o) s[tid] += s[tid + o];
        __syncthreads();
    }
    float denom = s[0];
    if (fabsf(denom) < 1e-10f) denom = 1e-10f;
    float u = (dxi - Binv_dg) / denom;
    if (conv) { u = 0.0f; vT = 0.0f; }
    Us[base * mi + k] = u;
    VTs[((long long)b * mi + k) * n + tid] = vT;
    if (!conv) {
        t[base] = tnew[base];
        g[base] = gni;
    }
}

// logdet via in-block Gaussian elimination (V SPD at the root -> no pivoting).
// W is scratch (reuses T buffer).  grid(B) x block(n)
__global__ void logdet_kernel(const float* __restrict__ V, float* __restrict__ W,
                              float* __restrict__ ld, int n) {
    int b = blockIdx.x, tid = threadIdx.x;
    const float* Vb = V + (long long)b * n * n;
    float* Wb = W + (long long)b * n * n;
    for (int idx = tid; idx < n * n; idx += blockDim.x) Wb[idx] = Vb[idx];
    __syncthreads();
    __shared__ float prow[N_SPINS];
    __shared__ float accs;
    if (tid == 0) accs = 0.0f;
    for (int k = 0; k < n; ++k) {
        __syncthreads();
        prow[tid] = Wb[(long long)k * n + tid];
        __syncthreads();
        if (tid == 0) accs += logf(fabsf(prow[k]));
        float piv = prow[k];
        if (tid > k) {
            float fct = Wb[(long long)tid * n + k] / piv;
            for (int j = k + 1; j < n; ++j)
                Wb[(long long)tid * n + j] -= fct * prow[j];
        }
    }
    __syncthreads();
    if (tid == 0) ld[b] = accs;
}

// phi[b] = sum_i t_i - 0.5*logdet + (1/256) * sum_{i,f} h*A.  grid(B) x block(n)
__global__ void phi_kernel(const float* __restrict__ t, const float* __restrict__ x,
                           const float* __restrict__ Aw, const float* __restrict__ ld,
                           float* __restrict__ out, int n, int f) {
    int b = blockIdx.x, tid = threadIdx.x;
    __shared__ float s[N_SPINS];
    const float* hr = x + ((long long)b * n + tid) * f;
    const float* ar = Aw + ((long long)b * n + tid) * f;
    float q = 0.0f;
    for (int j = 0; j < f; ++j) q += hr[j] * ar[j];
    s[tid] = t[(long long)b * n + tid] + q * (1.0f / 256.0f);
    __syncthreads();
    for (int o = N_SPINS / 2; o > 0; o >>= 1) {
        if (tid < o) s[tid] += s[tid + o];
        __syncthreads();
    }
    if (tid == 0) out[b] = s[0] - 0.5f * ld[b];
}

// ---------------------------------------------------------------------------
extern "C" void kernel_launch(void* const* d_in, const int* in_sizes, int n_in,
                              void* d_out, int out_size, void* d_ws, size_t ws_size,
                              hipStream_t stream) {
    (void)in_sizes; (void)n_in; (void)out_size; (void)ws_size;
    const int n = N_SPINS, f = F_DIM, b = B_BATCH, mi = MAX_IT;
    const float* x = (const float*)d_in[0];     // (B, N, F)
    const float* Jraw = (const float*)d_in[1];  // (N, N)

    char* base = (char*)d_ws;
    size_t off = 0;
    auto alloc = [&](size_t nf) -> float* {
        float* p = (float*)(base + off);
        off += nf * sizeof(float);
        off = (off + 255) & ~(size_t)255;
        return p;
    };
    float* J   = alloc((size_t)n * n);
    float* rs  = alloc((size_t)b * n);
    float* al  = alloc(b);
    float* V   = alloc((size_t)b * n * n);
    float* X   = alloc((size_t)b * n * n);
    float* X2  = alloc((size_t)b * n * n);
    float* T   = alloc((size_t)b * n * n);
    float* Aw  = alloc((size_t)b * n * f);
    float* t   = alloc((size_t)b * n);
    float* tn_ = alloc((size_t)b * n);
    float* g   = alloc((size_t)b * n);
    float* gn  = alloc((size_t)b * n);
    float* dx  = alloc((size_t)b * n);
    float* cv  = alloc(b);
    float* Us  = alloc((size_t)b * n * mi);
    float* VT  = alloc((size_t)b * mi * n);
    float* ld  = alloc(b);

    const long long snn = (long long)n * n;
    const long long snf = (long long)n * f;

    prep_J<<<dim3(n), dim3(n), 0, stream>>>(Jraw, J, n);
    init_state<<<dim3(b), dim3(n), 0, stream>>>(t, Us, VT, n, mi);

    auto eval_g = [&](const float* tcur, float* gout) {
        build_V<<<dim3(n, b), dim3(n), 0, stream>>>(J, tcur, V, rs, n);
        alpha_kernel<<<dim3(b), dim3(n), 0, stream>>>(rs, al, n);
        init_X<<<dim3(n, b), dim3(n), 0, stream>>>(al, X, n);
        float* Xc = X;
        float* Xo = X2;
        for (int it = 0; it < NS_IT; ++it) {
            // T = V @ Xc
            gemm_ns<<<dim3(16, b), dim3(128), 0, stream>>>(V, Xc, T, nullptr, 0);
            // Xo = 2*Xc - Xc @ T
            gemm_ns<<<dim3(16, b), dim3(128), 0, stream>>>(Xc, T, Xo, Xc, 1);
            float* tmp = Xc; Xc = Xo; Xo = tmp;
        }
        // NS_IT is even -> Xc == X holds V^-1
        // A = Vinv @ h   (M=256, N=64, K=256)
        gemm_wmma<<<dim3(16, b), dim3(128), 0, stream>>>(
            Xc, x, Aw, n, f, n, n, f, f, snn, snf, snf);
        grad_kernel<<<dim3(b), dim3(n), 0, stream>>>(Xc, Aw, gout, n, f);
    };

    eval_g(t, g);
    for (int k = 0; k < MAX_IT; ++k) {
        broyden_pre<<<dim3(b), dim3(n), 0, stream>>>(g, t, Us, VT, dx, tn_, cv, n, mi);
        eval_g(tn_, gn);
        broyden_post<<<dim3(b), dim3(n), 0, stream>>>(gn, g, t, tn_, dx, Us, VT, cv, k, n, mi);
    }
    // After the last step, V / X / Aw correspond to the final t (t_new == t when converged).
    logdet_kernel<<<dim3(b), dim3(n), 0, stream>>>(V, T, ld, n);
    phi_kernel<<<dim3(b), dim3(n), 0, stream>>>(t, x, Aw, ld, (float*)d_out, n, f);
}